// CNF_50285477101699
// MI455X (gfx1250) — compile-verified
//
#include <hip/hip_runtime.h>
#include <math.h>

// CNF exact-trace log-prob on gfx1250.
// fp32 WMMA (V_WMMA_F32_16X16X4_F32) for full-precision GEMM tiles.
// Weights pre-swizzled in LDS into WMMA A-operand order (bank-conflict-free,
// one b64 load per WMMA). Activations/gates via hardware v_tanh_f32.

typedef __attribute__((ext_vector_type(2))) float v2f;
typedef __attribute__((ext_vector_type(8))) float v8f;

#define LAT 16      // num latents
#define CND 16      // num conds
#define WID 128     // hidden width
#define NSTEPS 8
#define LOG2PI 1.8378770664093453f

// total dynamic LDS floats (see partition below)
#define SMEM_FLOATS 37232

__device__ __forceinline__ float fast_tanhf(float x) {
#if __has_builtin(__builtin_amdgcn_tanhf)
  return __builtin_amdgcn_tanhf(x);        // v_tanh_f32
#else
  return tanhf(x);
#endif
}

__device__ __forceinline__ float fast_sigmoid(float x) {
  // sigmoid(x) = 0.5*tanh(x/2) + 0.5  -> 1 TRANS + 2 VALU
  return fmaf(0.5f, fast_tanhf(0.5f * x), 0.5f);
}

__device__ __forceinline__ v8f wmma4(v2f a, v2f b, v8f c) {
  // D(16x16,f32) = A(16x4,f32) x B(4x16,f32) + C
  return __builtin_amdgcn_wmma_f32_16x16x4_f32(
      /*neg_a=*/false, a, /*neg_b=*/false, b,
      /*c_mod=*/(short)0, c, /*reuse_a=*/false, /*reuse_b=*/false);
}

__global__ __launch_bounds__(256) void cnf_logp_kernel(
    const float* __restrict__ W1_0, const float* __restrict__ b1_0,
    const float* __restrict__ W2_0, const float* __restrict__ b2_0,
    const float* __restrict__ W3_0,
    const float* __restrict__ W1_1, const float* __restrict__ b1_1,
    const float* __restrict__ W2_1, const float* __restrict__ b2_1,
    const float* __restrict__ W3_1,
    const float* __restrict__ W1_2, const float* __restrict__ b1_2,
    const float* __restrict__ W2_2, const float* __restrict__ b2_2,
    const float* __restrict__ W3_2,
    const float* __restrict__ z, const float* __restrict__ cond,
    float* __restrict__ out)
{
  extern __shared__ float smem[];
  // --- weights pre-swizzled into WMMA A-operand order [tile][kstep][lane]{2}
  float* sW10A = smem;                    // 8*8*32*2   = 4096  : L0 fwd A
  float* sW11A = sW10A + 4096;            // 8*32*32*2  = 16384 : L1 fwd + tangent A
  float* sW12A = sW11A + 16384;           // 32*32*2    = 2048  : L2 fwd A
  float* sW10T = sW12A + 2048;            // 32*32*2    = 2048  : tangent-B source W1_0[k][j]
  float* sW12R = sW10T + 2048;            // 8*8*32     = 2048  : trace reduce W1_2[j][kp]
  // --- small per-row vectors
  float* sb1_0 = sW12R + 2048;            // 128
  float* sb1_1 = sb1_0 + WID;             // 128
  float* sb1_2 = sb1_1 + WID;             // 16
  float* sW2_0 = sb1_2 + LAT;             // 128
  float* sb2_0 = sW2_0 + WID;             // 128
  float* sW3_0 = sb2_0 + WID;             // 128
  float* sW2_1 = sW3_0 + WID;             // 128
  float* sb2_1 = sW2_1 + WID;             // 128
  float* sW3_1 = sb2_1 + WID;             // 128
  float* sW2_2 = sW3_1 + WID;             // 16
  float* sb2_2 = sW2_2 + LAT;             // 16
  float* sW3_2 = sb2_2 + LAT;             // 16
  // --- activations / state
  float* sX    = sW3_2 + LAT;             // 32*16  : layer0 input [k][n]
  float* sH0   = sX  + (LAT+CND)*16;      // 128*16 : tanh activations layer0
  float* sA1   = sH0 + WID*16;            // 128*16 : (1-h0^2)*g0
  float* sH1   = sA1 + WID*16;            // 128*16
  float* sA2   = sH1 + WID*16;            // 128*16 : (1-h1^2)*g1
  float* sF    = sA2 + WID*16;            // 16*16  : vector field f [j][n]
  float* sDiv  = sF  + LAT*16;            // 16     : divergence per sample
  float* sY    = sDiv + 16;               // 16*16  : RK4 base state
  float* sYacc = sY  + LAT*16;            // 16*16
  float* sDacc = sYacc + LAT*16;          // 16
  float* sDlp  = sDacc + 16;              // 16

  const int tid  = threadIdx.x;
  const int lane = tid & 31;
  const int wv   = tid >> 5;              // wave id 0..7
  const int s0   = blockIdx.x * 16;       // first sample of this block

  // ---- stage weights into LDS, swizzled to WMMA operand layouts ----
  // A layout (f32 16x4): lane<16 -> row=M=lane, cols K,K+1; lane>=16 -> row=lane-16, cols K+2,K+3
  for (int idx = tid; idx < 8*32*32; idx += 256) {   // sW11A
    const int ln = idx & 31, kk = (idx >> 5) & 31, m = idx >> 10;
    const int row = 16*m + (ln & 15);
    const int col = 4*kk + ((ln < 16) ? 0 : 2);
    sW11A[idx*2+0] = W1_1[row*WID + col];
    sW11A[idx*2+1] = W1_1[row*WID + col + 1];
  }
  for (int idx = tid; idx < 8*8*32; idx += 256) {    // sW10A
    const int ln = idx & 31, kk = (idx >> 5) & 7, m = idx >> 8;
    const int row = 16*m + (ln & 15);
    const int col = 4*kk + ((ln < 16) ? 0 : 2);
    sW10A[idx*2+0] = W1_0[row*(LAT+CND) + col];
    sW10A[idx*2+1] = W1_0[row*(LAT+CND) + col + 1];
  }
  for (int idx = tid; idx < 32*32; idx += 256) {     // sW12A
    const int ln = idx & 31, kk = idx >> 5;
    const int row = ln & 15;
    const int col = 4*kk + ((ln < 16) ? 0 : 2);
    sW12A[idx*2+0] = W1_2[row*WID + col];
    sW12A[idx*2+1] = W1_2[row*WID + col + 1];
  }
  for (int idx = tid; idx < 32*32; idx += 256) {     // sW10T : B[k][j]=W1_0[k][j]
    const int ln = idx & 31, kk = idx >> 5;
    const int j = ln & 15;
    const int k = 4*kk + ((ln < 16) ? 0 : 2);
    sW10T[idx*2+0] = W1_0[k*(LAT+CND) + j];
    sW10T[idx*2+1] = W1_0[(k+1)*(LAT+CND) + j];
  }
  for (int idx = tid; idx < 8*8*32; idx += 256) {    // sW12R : W1_2[j][kp]
    const int ln = idx & 31, r = (idx >> 5) & 7, m = idx >> 8;
    const int j = ln & 15;
    const int kp = 16*m + r + ((ln < 16) ? 0 : 8);
    sW12R[idx] = W1_2[j*WID + kp];
  }
  if (tid < WID) {
    sb1_0[tid] = b1_0[tid]; sb1_1[tid] = b1_1[tid];
    sW2_0[tid] = W2_0[tid]; sb2_0[tid] = b2_0[tid]; sW3_0[tid] = W3_0[tid];
    sW2_1[tid] = W2_1[tid]; sb2_1[tid] = b2_1[tid]; sW3_1[tid] = W3_1[tid];
  }
  if (tid < LAT) {
    sb1_2[tid] = b1_2[tid];
    sW2_2[tid] = W2_2[tid]; sb2_2[tid] = b2_2[tid]; sW3_2[tid] = W3_2[tid];
  }
  // ---- state init: sample n = tid>>4, component j = tid&15 ----
  {
    const int n = tid >> 4, j = tid & 15;
    float yv = z[(s0 + n) * LAT + j];
    sY[j*16 + n]         = yv;
    sX[j*16 + n]         = yv;
    sX[(LAT + j)*16 + n] = cond[(s0 + n) * CND + j];
    if (j == 0) sDlp[n] = 0.0f;
  }
  __syncthreads();

  const float dt = -1.0f / NSTEPS;   // integrate t: 1 -> 0
  const int an   = lane & 15;        // B/C/D col (N) index for this lane
  const int koff = (lane < 16) ? 0 : 2;
  const int rhi  = (lane < 16) ? 0 : 8;
  const int l2   = lane * 2;         // per-lane offset into swizzled A tables

  for (int step = 0; step < NSTEPS; ++step) {
    const float tbase = 1.0f + step * dt;
    {
      const int n = tid >> 4, j = tid & 15;
      sYacc[j*16 + n] = 0.0f;
      if (j == 0) sDacc[n] = 0.0f;
    }
    __syncthreads();

    for (int stage = 0; stage < 4; ++stage) {
      const float t = tbase + ((stage == 1 || stage == 2) ? 0.5f * dt
                               : (stage == 3 ? dt : 0.0f));

      // ===== layer 0 forward (M-tile = wv, K = 32) =====
      v8f acc0 = {0.f,0.f,0.f,0.f,0.f,0.f,0.f,0.f};
      {
        const float* aW = &sW10A[wv * (8*64)];
#pragma unroll
        for (int kk = 0; kk < (LAT+CND)/4; ++kk) {
          const int kb = 4*kk + koff;
          v2f a = *(const v2f*)&aW[kk*64 + l2];
          v2f b; b.x = sX[kb*16 + an]; b.y = sX[(kb+1)*16 + an];
          acc0 = wmma4(a, b, acc0);
        }
      }
#pragma unroll
      for (int r = 0; r < 8; ++r) {
        const int row = 16*wv + r + rhi;
        const float g = fast_sigmoid(sW2_0[row]*t + sb2_0[row]);
        const float u = (acc0[r] + sb1_0[row])*g + sW3_0[row]*t;
        const float h = fast_tanhf(u);
        sH0[row*16 + an] = h;
        sA1[row*16 + an] = (1.0f - h*h)*g;
      }
      __syncthreads();

      // ===== layer 1 forward (M-tile = wv, K = 128) =====
      v8f acc1 = {0.f,0.f,0.f,0.f,0.f,0.f,0.f,0.f};
      {
        const float* aW = &sW11A[wv * (32*64)];
#pragma unroll
        for (int kk = 0; kk < WID/4; ++kk) {
          const int kb = 4*kk + koff;
          v2f a = *(const v2f*)&aW[kk*64 + l2];
          v2f b; b.x = sH0[kb*16 + an]; b.y = sH0[(kb+1)*16 + an];
          acc1 = wmma4(a, b, acc1);
        }
      }
#pragma unroll
      for (int r = 0; r < 8; ++r) {
        const int row = 16*wv + r + rhi;
        const float g = fast_sigmoid(sW2_1[row]*t + sb2_1[row]);
        const float u = (acc1[r] + sb1_1[row])*g + sW3_1[row]*t;
        const float h = fast_tanhf(u);
        sH1[row*16 + an] = h;
        sA2[row*16 + an] = (1.0f - h*h)*g;
      }
      __syncthreads();

      // ===== layer 2 forward (wave 0 only; M=16, K=128) =====
      if (wv == 0) {
        v8f acc2 = {0.f,0.f,0.f,0.f,0.f,0.f,0.f,0.f};
#pragma unroll
        for (int kk = 0; kk < WID/4; ++kk) {
          const int kb = 4*kk + koff;
          v2f a = *(const v2f*)&sW12A[kk*64 + l2];
          v2f b; b.x = sH1[kb*16 + an]; b.y = sH1[(kb+1)*16 + an];
          acc2 = wmma4(a, b, acc2);
        }
#pragma unroll
        for (int r = 0; r < 8; ++r) {
          const int row = r + rhi;
          const float g = fast_sigmoid(sW2_2[row]*t + sb2_2[row]);
          sF[row*16 + an] = (acc2[r] + sb1_2[row])*g + sW3_2[row]*t;
        }
      }

      // ===== exact-trace tangents: wave wv owns samples 2wv, 2wv+1 =====
      // T1_s[k][j] = a1_s[k] * W1_0[k][j]  (layer-0 tangent, seed [I;0])
      // U = W1_1 @ T1_s ; div_s = g2[j] * sum_{j,k} W1_2[j,k]*a2_s[k]*U[k,j]
      const float g2j = fast_sigmoid(sW2_2[an]*t + sb2_2[an]);   // j = an
#pragma unroll
      for (int sl = 0; sl < 2; ++sl) {
        const int s = 2*wv + sl;
        // build B operand once per sample: 32 x v2f in registers
        v2f bc[32];
#pragma unroll
        for (int kk = 0; kk < WID/4; ++kk) {
          const int kb = 4*kk + koff;
          v2f wt = *(const v2f*)&sW10T[kk*64 + l2];
          bc[kk].x = sA1[kb*16 + s]     * wt.x;
          bc[kk].y = sA1[(kb+1)*16 + s] * wt.y;
        }
        float divacc = 0.0f;
        for (int m = 0; m < 8; ++m) {          // M-tiles of U (rows k)
          v8f accT = {0.f,0.f,0.f,0.f,0.f,0.f,0.f,0.f};
          const float* aW = &sW11A[m * (32*64)];
#pragma unroll
          for (int kk = 0; kk < WID/4; ++kk) { // K = 128: 1 b64 load + 1 wmma
            v2f a = *(const v2f*)&aW[kk*64 + l2];
            accT = wmma4(a, bc[kk], accT);
          }
          const float* wR = &sW12R[m * 256 + lane];
#pragma unroll
          for (int r = 0; r < 8; ++r) {
            const int kp = 16*m + r + rhi;     // row of U; col of U = an
            divacc += wR[r*32] * sA2[kp*16 + s] * accT[r];
          }
        }
        divacc *= g2j;
        for (int off = 16; off >= 1; off >>= 1)
          divacc += __shfl_xor(divacc, off, 32);
        if (lane == 0) sDiv[s] = divacc;
      }
      __syncthreads();

      // ===== RK4 bookkeeping (per sample n, component j) =====
      {
        const int n = tid >> 4, j = tid & 15;
        const float fv  = sF[j*16 + n];
        const float wgt = (stage == 0 || stage == 3) ? 1.0f : 2.0f;
        sYacc[j*16 + n] += wgt * fv;
        if (j == 0) sDacc[n] += wgt * sDiv[n];
        if (stage < 3) {
          const float coef = (stage == 2) ? dt : 0.5f * dt;
          sX[j*16 + n] = sY[j*16 + n] + coef * fv;
        } else {
          const float ynew = sY[j*16 + n] + (dt / 6.0f) * sYacc[j*16 + n];
          sY[j*16 + n] = ynew;
          sX[j*16 + n] = ynew;                 // stage-0 input of next step
          if (j == 0) sDlp[n] += (dt / 6.0f) * sDacc[n];
        }
      }
      __syncthreads();
    }
  }

  // ===== final: delta_logp + standard-normal log-prob =====
  if (tid < 16) {
    float acc = 0.0f;
    for (int j = 0; j < LAT; ++j) {
      const float yv = sY[j*16 + tid];
      acc += -0.5f * yv * yv - 0.5f * LOG2PI;
    }
    out[s0 + tid] = sDlp[tid] + acc;
  }
}

extern "C" void kernel_launch(void* const* d_in, const int* in_sizes, int n_in,
                              void* d_out, int out_size, void* d_ws, size_t ws_size,
                              hipStream_t stream) {
  (void)n_in; (void)d_ws; (void)ws_size; (void)out_size;
  const float* W1_0 = (const float*)d_in[0];
  const float* b1_0 = (const float*)d_in[1];
  const float* W2_0 = (const float*)d_in[2];
  const float* b2_0 = (const float*)d_in[3];
  const float* W3_0 = (const float*)d_in[4];
  const float* W1_1 = (const float*)d_in[5];
  const float* b1_1 = (const float*)d_in[6];
  const float* W2_1 = (const float*)d_in[7];
  const float* b2_1 = (const float*)d_in[8];
  const float* W3_1 = (const float*)d_in[9];
  const float* W1_2 = (const float*)d_in[10];
  const float* b1_2 = (const float*)d_in[11];
  const float* W2_2 = (const float*)d_in[12];
  const float* b2_2 = (const float*)d_in[13];
  const float* W3_2 = (const float*)d_in[14];
  const float* z    = (const float*)d_in[15];
  const float* cnd  = (const float*)d_in[16];
  float* out = (float*)d_out;

  const int B = in_sizes[15] / LAT;        // 2048
  const int nblocks = B / 16;              // 128 blocks of 16 samples
  const size_t shmem = SMEM_FLOATS * sizeof(float);  // ~145.4 KB

  (void)hipFuncSetAttribute((const void*)cnf_logp_kernel,
                            hipFuncAttributeMaxDynamicSharedMemorySize,
                            (int)shmem);
  cnf_logp_kernel<<<nblocks, 256, shmem, stream>>>(
      W1_0, b1_0, W2_0, b2_0, W3_0,
      W1_1, b1_1, W2_1, b2_1, W3_1,
      W1_2, b1_2, W2_2, b2_2, W3_2,
      z, cnd, out);
}